// EvolutionModel_57947698757730
// MI455X (gfx1250) — compile-verified
//
#include <hip/hip_runtime.h>
#include <math.h>

typedef float v2f __attribute__((ext_vector_type(2)));
typedef float v8f __attribute__((ext_vector_type(8)));

#define TWO_PI_F 6.28318530717958647692f
#define NSTEPS 16
#define HIST 17
#define EV_BLOCK 128
#define RAY_GROUPS 4   // 16-ray WMMA column groups per wave

// ---------------------------------------------------------------------------
// Kernel 1: build the homogeneous barycentric A-matrix for the WMMA search.
// Row layout per tet t (4 rows x 4 cols, row-major):
//   rows 0..2:  [ ort[t][i][0..2] , -dot(ort[t][i], origin[t]) ]   -> lam_i
//   row  3   :  column-wise sum of rows 0..2                        -> sum(lam)
// Padding tets (t >= T) are filled so they can never test valid.
// ---------------------------------------------------------------------------
__global__ void build_amat_kernel(const float* __restrict__ ort,
                                  const float* __restrict__ origin,
                                  float* __restrict__ Amat,
                                  int T, int nTetPad) {
  int t = blockIdx.x * blockDim.x + threadIdx.x;
  if (t >= nTetPad) return;
  float* row = Amat + (size_t)t * 16;
  if (t < T) {
    float o0 = origin[t*3+0], o1 = origin[t*3+1], o2 = origin[t*3+2];
    float s0 = 0.f, s1 = 0.f, s2 = 0.f, s3 = 0.f;
#pragma unroll
    for (int i = 0; i < 3; ++i) {
      float w0 = ort[t*9 + i*3 + 0];
      float w1 = ort[t*9 + i*3 + 1];
      float w2 = ort[t*9 + i*3 + 2];
      float c  = -(w0*o0 + w1*o1 + w2*o2);
      row[i*4+0] = w0; row[i*4+1] = w1; row[i*4+2] = w2; row[i*4+3] = c;
      s0 += w0; s1 += w1; s2 += w2; s3 += c;
    }
    row[12] = s0; row[13] = s1; row[14] = s2; row[15] = s3;
  } else {
#pragma unroll
    for (int i = 0; i < 3; ++i) {
      row[i*4+0] = 0.f; row[i*4+1] = 0.f; row[i*4+2] = 0.f; row[i*4+3] = -1.f;
    }
    row[12] = 0.f; row[13] = 0.f; row[14] = 0.f; row[15] = 2.f; // sum=2 > 1
  }
}

// ---------------------------------------------------------------------------
// Kernel 2: point-in-tet search via V_WMMA_F32_16X16X4_F32.
// One wave holds 64 rays as 4 B-operands (16 columns each); each A tile
// (16 rows = 4 tets) is loaded once per wave with a single b64 and feeds
// 4 independent back-to-back WMMAs -> load latency hidden, A traffic /4.
//   A 16x4 layout : lane l holds M = l&15 ; VGPR0 = K (0 | 2), VGPR1 = K (1 | 3)
//   B 4x16 layout : lane l holds N = l&15 ; VGPR0 = K (0 | 2), VGPR1 = K (1 | 3)
//   C 16x16       : VGPR v, lanes 0-15 -> row v, lanes 16-31 -> row v+8
// Lanes 0-15 judge tets {0,1} of the tile, lanes 16-31 tets {2,3}, for the
// same ray column; shfl_xor(16) merges halves; max valid index == reference's
// "last valid tet or -1".
// ---------------------------------------------------------------------------
__global__ void find_tet_kernel(const float* __restrict__ r0,
                                const float* __restrict__ Amat,
                                int* __restrict__ tet0,
                                int nTiles, int T, int B) {
  const int lane = threadIdx.x & 31;
  const int wave = threadIdx.x >> 5;
  const int group = blockIdx.x * (blockDim.x >> 5) + wave;
  const int rayBase = group * (16 * RAY_GROUPS);
  if (rayBase >= B) return;                 // uniform per wave, EXEC stays full
  const int n  = lane & 15;
  const bool hi = lane >= 16;

  v2f bvec[RAY_GROUPS];
  int best[RAY_GROUPS];
#pragma unroll
  for (int g = 0; g < RAY_GROUPS; ++g) {
    int ray  = rayBase + g*16 + n;
    int rayC = ray < B ? ray : B - 1;
    float px = r0[rayC*3+0], py = r0[rayC*3+1], pz = r0[rayC*3+2];
    bvec[g].x = hi ? pz : px;               // K=0(x) | K=2(z)
    bvec[g].y = hi ? 1.0f : py;             // K=1(y) | K=3(1)
    best[g] = -1;
  }

  const float* aptr = Amat + (size_t)n*4 + (hi ? 2 : 0);
  for (int tile = 0; tile < nTiles; ++tile) {
    v2f avec = *(const v2f*)(aptr + (size_t)tile*64);   // one b64 per lane
    int pf = (tile + 2 < nTiles) ? tile + 2 : tile;
    __builtin_prefetch(aptr + (size_t)pf*64, 0, 3);
    const int tetBase = tile*4 + (hi ? 2 : 0);
#pragma unroll
    for (int g = 0; g < RAY_GROUPS; ++g) {
      v8f acc = {};
      acc = __builtin_amdgcn_wmma_f32_16x16x4_f32(
          /*neg_a=*/false, avec, /*neg_b=*/false, bvec[g],
          /*c_mod=*/(short)0, acc, /*reuse_a=*/false, /*reuse_b=*/false);
#pragma unroll
      for (int j = 0; j < 2; ++j) {
        float l0 = acc[4*j+0], l1 = acc[4*j+1], l2 = acc[4*j+2], ls = acc[4*j+3];
        float lmin = fminf(fminf(l0, l1), l2);   // -> v_min3_f32
        float lmax = fmaxf(fmaxf(l0, l1), l2);   // -> v_max3_f32
        bool valid = (lmin >= 0.f) && (lmax <= 1.f) && (ls <= 1.f);
        int tt = tetBase + j;
        best[g] = (valid && tt < T && tt > best[g]) ? tt : best[g];
      }
    }
  }
#pragma unroll
  for (int g = 0; g < RAY_GROUPS; ++g) {
    int other = __shfl_xor(best[g], 16, 32);    // merge lane halves (wave32)
    int b2 = best[g] > other ? best[g] : other;
    int ray = rayBase + g*16 + n;
    if (!hi && ray < B) tet0[ray] = b2;
  }
}

// ---------------------------------------------------------------------------
// Kernel 3: 16 arc-evolution steps + depth interpolation per ray.
// History (17 positions + cumulative distances) lives in LDS so the
// data-dependent final gather is ds_load, not scratch.
// ---------------------------------------------------------------------------
__global__ void evolve_sample_kernel(const float* __restrict__ r0,
                                     const float* __restrict__ m0,
                                     const float* __restrict__ z_vals,
                                     const float* __restrict__ pos,
                                     const float* __restrict__ a_t,
                                     const float* __restrict__ b_t,
                                     const float* __restrict__ n_t,
                                     const int* __restrict__ tetra_face,
                                     const int* __restrict__ face_vertex,
                                     const int* __restrict__ face_tetra,
                                     const int* __restrict__ tet0,
                                     float* __restrict__ out,
                                     int B, int nz) {
  __shared__ float shx[EV_BLOCK*HIST];
  __shared__ float shy[EV_BLOCK*HIST];
  __shared__ float shz[EV_BLOCK*HIST];
  __shared__ float shd[EV_BLOCK*HIST];

  const int ray = blockIdx.x * blockDim.x + threadIdx.x;
  if (ray >= B) return;
  const int tid = threadIdx.x;

  float rx = r0[ray*3+0], ry = r0[ray*3+1], rz = r0[ray*3+2];
  float mx = m0[ray*3+0], my = m0[ray*3+1], mz = m0[ray*3+2];
  int tidx = tet0[ray];

  shx[tid*HIST] = rx; shy[tid*HIST] = ry; shz[tid*HIST] = rz; shd[tid*HIST] = 0.f;
  float cum = 0.f;

  for (int step = 0; step < NSTEPS; ++step) {
    int ti = tidx > 0 ? tidx : 0;                       // max(tidx, 0)
    float av = a_t[ti];
    float bx = b_t[ti*3+0], by = b_t[ti*3+1], bz = b_t[ti*3+2];
    float nx = n_t[ti*3+0], ny = n_t[ti*3+1], nz_ = n_t[ti*3+2];
    int faces[4];
#pragma unroll
    for (int f = 0; f < 4; ++f) faces[f] = tetra_face[ti*4+f];

    // circle center rc and radius vector R
    float mnx = my*nz_ - mz*ny;
    float mny = mz*nx  - mx*nz_;
    float mnz = mx*ny  - my*nx;
    float mnn = sqrtf(mnx*mnx + mny*mny + mnz*mnz);
    float qx = mnx/mnn, qy = mny/mnn, qz = mnz/mnn;
    float nqx = ny*qz - nz_*qy;
    float nqy = nz_*qx - nx*qz;
    float nqz = nx*qy - ny*qx;
    float mndot = mx*nx + my*ny + mz*nz_;
    float bn   = sqrtf(bx*bx + by*by + bz*bz);
    float coef = (rx*nx + ry*ny + rz*nz_) + av/bn;
    float mdnq = mx*nqx + my*nqy + mz*nqz;
    float g = mndot/mdnq;
    float wx = nx - g*nqx, wy = ny - g*nqy, wz = nz_ - g*nqz;
    float rcx = rx - coef*wx, rcy = ry - coef*wy, rcz = rz - coef*wz;
    float Rx = rcx - rx, Ry = rcy - ry, Rz = rcz - rz;
    float Rn = sqrtf(Rx*Rx + Ry*Ry + Rz*Rz);

    // exit-face search (argmin of exit angle, first-tie like jnp.argmin)
    float bestPhi = 1e30f; int bestF = 0;
#pragma unroll
    for (int f = 0; f < 4; ++f) {
      int fid = faces[f];
      int vi = face_vertex[fid*3+0], vj = face_vertex[fid*3+1], vk = face_vertex[fid*3+2];
      float ix = pos[vi*3+0], iy = pos[vi*3+1], iz = pos[vi*3+2];
      float jx = pos[vj*3+0], jy = pos[vj*3+1], jz = pos[vj*3+2];
      float kx = pos[vk*3+0], ky = pos[vk*3+1], kz = pos[vk*3+2];
      float e1x = jx-ix, e1y = jy-iy, e1z = jz-iz;
      float e2x = kx-ix, e2y = ky-iy, e2z = kz-iz;
      float MLx = e1y*e2z - e1z*e2y;
      float MLy = e1z*e2x - e1x*e2z;
      float MLz = e1x*e2y - e1y*e2x;
      float QL = -(ix*MLx + iy*MLy + iz*MLz);
      float c1 = -(MLx*Rx + MLy*Ry + MLz*Rz);
      float c2 = Rn*(MLx*mx + MLy*my + MLz*mz);
      float c3 = (MLx*rcx + MLy*rcy + MLz*rcz) + QL;
      float disc = c1*c1 + c2*c2 - c3*c3;
      bool ok = disc >= 0.f;
      float sq = sqrtf(ok ? disc : 0.f);
      float den = c1 - c3;
      den = (den == 0.f) ? 1e-20f : den;
      float p1 = fmodf(2.f*atanf((c2+sq)/den), TWO_PI_F); p1 += (p1 < 0.f) ? TWO_PI_F : 0.f;
      float p2 = fmodf(2.f*atanf((c2-sq)/den), TWO_PI_F); p2 += (p2 < 0.f) ? TWO_PI_F : 0.f;
      float phiR = ok ? fminf(p1, p2) : 10.f;
      if (phiR < bestPhi) { bestPhi = phiR; bestF = f; }
    }

    float phiE = bestPhi * 1.01f;
    float cE = cosf(phiE), sE = sinf(phiE);
    float rex = rcx - cE*Rx + Rn*sE*mx;
    float rey = rcy - cE*Ry + Rn*sE*my;
    float rez = rcz - cE*Rz + Rn*sE*mz;
    float inR = sE / Rn;
    float mex = cE*mx + inR*Rx;
    float mey = cE*my + inR*Ry;
    float mez = cE*mz + inR*Rz;

    int hit = faces[bestF];
    int f0 = face_tetra[hit*2+0], f1 = face_tetra[hit*2+1];
    int nxt = (f0 == tidx) ? f1 : f0;
    nxt = (tidx < 0) ? -1 : nxt;

    float dxx = rx - rex, dyy = ry - rey, dzz = rz - rez;
    cum += sqrtf(dxx*dxx + dyy*dyy + dzz*dzz);

    shx[tid*HIST+step+1] = rex;
    shy[tid*HIST+step+1] = rey;
    shz[tid*HIST+step+1] = rez;
    shd[tid*HIST+step+1] = cum;

    rx = rex; ry = rey; rz = rez;
    mx = mex; my = mey; mz = mez;
    tidx = nxt;
  }

  // depth sampling: bracket each z between history distances, interpolate
  for (int zi = 0; zi < nz; ++zi) {
    float z = z_vals[ray*nz + zi];
    float minPos = 1e30f, maxNeg = -1e30f;
    int i0 = 0, i1 = 0;
    for (int k = 0; k < HIST; ++k) {
      float tm = z - shd[tid*HIST+k];
      float tp = (tm < 0.f) ? 10.f  : tm;
      float tn = (tm > 0.f) ? -10.f : tm;
      if (tp < minPos) { minPos = tp; i0 = k; }
      if (tn > maxNeg) { maxNeg = tn; i1 = k; }
    }
    float c0x = shx[tid*HIST+i0], c0y = shy[tid*HIST+i0], c0z = shz[tid*HIST+i0];
    float c1x = shx[tid*HIST+i1], c1y = shy[tid*HIST+i1], c1z = shz[tid*HIST+i1];
    float ddx = (c1x-c0x)/z, ddy = (c1y-c0y)/z, ddz = (c1z-c0z)/z;
    float dn = sqrtf(ddx*ddx + ddy*ddy + ddz*ddz);
    float ux = ddx/dn, uy = ddy/dn, uz = ddz/dn;
    size_t o = ((size_t)ray*nz + zi)*3;
    out[o+0] = c0x + minPos*ux;
    out[o+1] = c0y + minPos*uy;
    out[o+2] = c0z + minPos*uz;
  }
}

// ---------------------------------------------------------------------------
extern "C" void kernel_launch(void* const* d_in, const int* in_sizes, int n_in,
                              void* d_out, int out_size, void* d_ws, size_t ws_size,
                              hipStream_t stream) {
  const float* r0          = (const float*)d_in[0];
  const float* m0          = (const float*)d_in[1];
  const float* z_vals      = (const float*)d_in[2];
  const float* pos         = (const float*)d_in[3];
  const float* a           = (const float*)d_in[4];
  const float* b           = (const float*)d_in[5];
  const float* n           = (const float*)d_in[6];
  const float* origin      = (const float*)d_in[7];
  const float* ort         = (const float*)d_in[8];
  const int*   tetra_face  = (const int*)d_in[9];
  const int*   face_vertex = (const int*)d_in[10];
  const int*   face_tetra  = (const int*)d_in[11];

  const int B  = in_sizes[0] / 3;   // 8192
  const int T  = in_sizes[4];       // 3072
  const int nz = in_sizes[2] / B;   // 64

  const int nTetPad = ((T + 3) / 4) * 4;
  const int nTiles  = nTetPad / 4;  // 16 A-rows (= 4 tets) per tile

  float* Amat = (float*)d_ws;
  int*   tet0 = (int*)((char*)d_ws + (size_t)nTetPad * 16 * sizeof(float));

  {
    int threads = 256;
    int blocks  = (nTetPad + threads - 1) / threads;
    hipLaunchKernelGGL(build_amat_kernel, dim3(blocks), dim3(threads), 0, stream,
                       ort, origin, Amat, T, nTetPad);
  }
  {
    int threads = 256;                        // 8 waves per block
    int wavesPerBlock = threads / 32;
    int raysPerWave = 16 * RAY_GROUPS;        // 64
    int groups = (B + raysPerWave - 1) / raysPerWave;
    int blocks = (groups + wavesPerBlock - 1) / wavesPerBlock;
    hipLaunchKernelGGL(find_tet_kernel, dim3(blocks), dim3(threads), 0, stream,
                       r0, Amat, tet0, nTiles, T, B);
  }
  {
    int threads = EV_BLOCK;
    int blocks  = (B + threads - 1) / threads;
    hipLaunchKernelGGL(evolve_sample_kernel, dim3(blocks), dim3(threads), 0, stream,
                       r0, m0, z_vals, pos, a, b, n, tetra_face, face_vertex,
                       face_tetra, tet0, (float*)d_out, B, nz);
  }
}